// DetectPostProcess_75161927680371
// MI455X (gfx1250) — compile-verified
//
#include <hip/hip_runtime.h>
#include <hip/hip_bf16.h>

#define B_     32
#define N_     24564
#define C_     81
#define NCLS   80
#define NPAIR  (B_ * NCLS)      // 2560
#define TOPK   100
#define MAXCAP 4096
#define TH_CONF_ 0.05f
#define TH_IOU_  0.5f

typedef __attribute__((ext_vector_type(2))) float v2f;
typedef __attribute__((ext_vector_type(8))) float v8f;

// ---------------------------------------------------------------------------
// Kernel 0: zero the per-(b,class) candidate counters (graph-safe init).
// ---------------------------------------------------------------------------
__global__ void k_zero_counters(unsigned* __restrict__ counters) {
    int i = blockIdx.x * blockDim.x + threadIdx.x;
    if (i < NPAIR) counters[i] = 0u;
}

// ---------------------------------------------------------------------------
// Kernel 1: streaming softmax over conf (single pass over 255MB), WMMA-based
// row-sum of exp, threshold at 0.05, scatter candidates to global buckets.
// One wave handles 16 rows; lane layout matches V_WMMA_F32_16X16X4_F32 A-matrix:
//   lane m (m<16)  : row m, K = k0+0, k0+1
//   lane m+16      : row m, K = k0+2, k0+3
// B = all-ones 4x16  =>  D[m][*] = row-sum of the 4-wide chunk, C-accumulated.
// ---------------------------------------------------------------------------
__global__ void k_softmax_cand(const float* __restrict__ conf,
                               unsigned* __restrict__ counters,
                               float* __restrict__ cscore,
                               int* __restrict__ cidx,
                               int cap) {
    const int wave = threadIdx.x >> 5;
    const int lane = threadIdx.x & 31;
    const int half = lane >> 4;
    const int m    = lane & 15;
    const long row = (long)blockIdx.x * 128 + wave * 16 + m;   // grid exact
    const float* rp = conf + row * C_;

    // Load this lane's share of 16 rows x 80 classes (40 floats / lane).
    float a[40];
#pragma unroll
    for (int j = 0; j < 20; j++) {
        a[2 * j]     = rp[4 * j + 2 * half];
        a[2 * j + 1] = rp[4 * j + 2 * half + 1];
    }
    float x80 = (half == 0) ? rp[80] : -3.0e38f;   // 81st class on half0 lanes

    // Row max (lane-local, then combine the two half-lanes of each row).
    float mx = x80;
#pragma unroll
    for (int j = 0; j < 40; j++) mx = fmaxf(mx, a[j]);
    mx = fmaxf(mx, __shfl_xor(mx, 16, 32));

    // exp on the VALU trans pipe (co-executes with WMMA below).
#pragma unroll
    for (int j = 0; j < 40; j++) a[j] = __expf(a[j] - mx);
    const float e80 = (half == 0) ? __expf(x80 - mx) : 0.0f;

    // WMMA accumulation: 20 chunks of K=4 cover classes 0..79.
    v8f c = {};
    v2f bones = {1.0f, 1.0f};
#pragma unroll
    for (int j = 0; j < 20; j++) {
        v2f av;
        av.x = a[2 * j];
        av.y = a[2 * j + 1];
        c = __builtin_amdgcn_wmma_f32_16x16x4_f32(false, av, false, bones,
                                                  (short)0, c, false, false);
    }

    // D layout: VGPR v, lanes 0-15 -> row v; lanes 16-31 -> row v+8.
    __shared__ float sums[8][16];
    if (lane == 0) {
#pragma unroll
        for (int i = 0; i < 8; i++) sums[wave][i] = c[i];
    }
    if (lane == 16) {
#pragma unroll
        for (int i = 0; i < 8; i++) sums[wave][8 + i] = c[i];
    }
    __syncthreads();

    float inv = 0.0f;
    if (half == 0) inv = 1.0f / (sums[wave][m] + e80);
    inv = __shfl(inv, m, 32);   // broadcast half0 -> both halves of each row

    // Threshold + scatter candidates (classes 1..80 only; class0 is background).
    const int b = (int)(row / N_);
    const int n = (int)(row % N_);
#pragma unroll
    for (int j = 0; j < 20; j++) {
#pragma unroll
        for (int t = 0; t < 2; t++) {
            const int cls = 4 * j + 2 * half + t;
            const float s = a[2 * j + t] * inv;
            if (cls >= 1 && s >= TH_CONF_) {
                const int ci = b * NCLS + (cls - 1);
                const unsigned pos = atomicAdd(&counters[ci], 1u);
                if (pos < (unsigned)cap) {
                    const size_t off = (size_t)ci * (size_t)cap + pos;
                    cscore[off] = s;
                    cidx[off]   = n;
                }
            }
        }
    }
    if (half == 0) {
        const float s = e80 * inv;
        if (s >= TH_CONF_) {
            const int ci = b * NCLS + (80 - 1);
            const unsigned pos = atomicAdd(&counters[ci], 1u);
            if (pos < (unsigned)cap) {
                const size_t off = (size_t)ci * (size_t)cap + pos;
                cscore[off] = s;
                cidx[off]   = n;
            }
        }
    }
}

// ---------------------------------------------------------------------------
// Kernel 2: per (b,class) block — top-100 selection (argmax with lowest-index
// tie-break, matching lax.top_k), box decode of only the 100 winners, greedy
// NMS in LDS, masked output write.
// ---------------------------------------------------------------------------
__global__ void k_select_nms(const unsigned* __restrict__ counters,
                             const float* __restrict__ cscore,
                             const int* __restrict__ cidx,
                             const float* __restrict__ loc,
                             const float* __restrict__ anchors,
                             float* __restrict__ out,
                             int cap) {
    __shared__ float cs[MAXCAP];
    __shared__ int   cn[MAXCAP];
    __shared__ float rs[256];
    __shared__ int   rn[256];
    __shared__ int   rpp[256];
    __shared__ float tS[TOPK];
    __shared__ int   tN[TOPK];
    __shared__ float bx1[TOPK], by1[TOPK], bx2[TOPK], by2[TOPK];
    __shared__ float bar[TOPK], bsc[TOPK];
    __shared__ int   keep[TOPK];

    const int ci  = blockIdx.x;
    const int b   = ci / NCLS;
    const int tid = threadIdx.x;

    int count = (int)counters[ci];
    if (count > cap)    count = cap;
    if (count > MAXCAP) count = MAXCAP;

    for (int p = tid; p < count; p += 256) {
        const size_t off = (size_t)ci * (size_t)cap + p;
        cs[p] = cscore[off];
        cn[p] = cidx[off];
    }
    __syncthreads();

    // --- top-100 by repeated argmax (score desc, index asc on ties) ---
    for (int k = 0; k < TOPK; k++) {
        float bs = -1.0f;
        int bn = 0x7fffffff, bp = -1;
        for (int p = tid; p < count; p += 256) {
            const float s = cs[p];
            const int nn = cn[p];
            if (s > bs || (s == bs && nn < bn)) { bs = s; bn = nn; bp = p; }
        }
        rs[tid] = bs; rn[tid] = bn; rpp[tid] = bp;
        __syncthreads();
        for (int off = 128; off >= 1; off >>= 1) {
            if (tid < off) {
                const float s2 = rs[tid + off];
                const int   n2 = rn[tid + off];
                if (s2 > rs[tid] || (s2 == rs[tid] && n2 < rn[tid])) {
                    rs[tid] = s2; rn[tid] = n2; rpp[tid] = rpp[tid + off];
                }
            }
            __syncthreads();
        }
        if (tid == 0) {
            if (rs[0] > 0.0f) {
                tS[k] = rs[0]; tN[k] = rn[0];
                cs[rpp[0]] = -2.0f;                // mark consumed
            } else {
                tS[k] = 0.0f; tN[k] = -1;
            }
        }
        __syncthreads();
    }

    // --- decode only the winners ---
    if (tid < TOPK) {
        const int n = tN[tid];
        if (n >= 0) {
            const float4 l = *(const float4*)(loc + ((size_t)b * N_ + n) * 4);
            const float4 A = *(const float4*)(anchors + (size_t)n * 4);
            const float cx = A.x + l.x * 0.1f * A.z;
            const float cy = A.y + l.y * 0.1f * A.w;
            const float w  = A.z * __expf(l.z * 0.2f);
            const float h  = A.w * __expf(l.w * 0.2f);
            const float X1 = cx - 0.5f * w, Y1 = cy - 0.5f * h;
            const float X2 = cx + 0.5f * w, Y2 = cy + 0.5f * h;
            bx1[tid] = X1; by1[tid] = Y1; bx2[tid] = X2; by2[tid] = Y2;
            bar[tid] = fmaxf(X2 - X1, 0.0f) * fmaxf(Y2 - Y1, 0.0f);
            bsc[tid] = tS[tid];
            keep[tid] = 1;
        } else {
            bx1[tid] = by1[tid] = bx2[tid] = by2[tid] = 0.0f;
            bar[tid] = 0.0f; bsc[tid] = 0.0f;
            keep[tid] = 0;
        }
    }

    // --- greedy NMS (sequential over i, parallel over j) ---
    for (int i = 0; i < TOPK; i++) {
        __syncthreads();
        if (tid < TOPK && tid > i && keep[i] && keep[tid]) {
            const float xx1 = fmaxf(bx1[i], bx1[tid]);
            const float yy1 = fmaxf(by1[i], by1[tid]);
            const float xx2 = fminf(bx2[i], bx2[tid]);
            const float yy2 = fminf(by2[i], by2[tid]);
            const float iw = fmaxf(xx2 - xx1, 0.0f);
            const float ih = fmaxf(yy2 - yy1, 0.0f);
            const float inter = iw * ih;
            const float iou = inter / (bar[i] + bar[tid] - inter + 1e-12f);
            if (iou > TH_IOU_) keep[tid] = 0;
        }
    }
    __syncthreads();

    if (tid < TOPK) {
        const float f = keep[tid] ? 1.0f : 0.0f;
        float* o = out + ((size_t)ci * TOPK + tid) * 5;
        o[0] = bx1[tid] * f;
        o[1] = by1[tid] * f;
        o[2] = bx2[tid] * f;
        o[3] = by2[tid] * f;
        o[4] = bsc[tid] * f;
    }
}

// ---------------------------------------------------------------------------
extern "C" void kernel_launch(void* const* d_in, const int* in_sizes, int n_in,
                              void* d_out, int out_size, void* d_ws, size_t ws_size,
                              hipStream_t stream) {
    const float* conf    = (const float*)d_in[0];
    const float* loc     = (const float*)d_in[1];
    const float* anchors = (const float*)d_in[2];
    float* out           = (float*)d_out;

    // Workspace layout: [counters: NPAIR u32][cscore: NPAIR*cap f32][cidx: NPAIR*cap i32]
    const size_t cbytes = (size_t)NPAIR * sizeof(unsigned);
    const size_t avail  = (ws_size > cbytes) ? (ws_size - cbytes) : 0;
    size_t cap_sz = avail / ((size_t)NPAIR * 8);
    int cap = (cap_sz > (size_t)MAXCAP) ? MAXCAP : (int)cap_sz;   // cap==0 is safe

    unsigned* counters = (unsigned*)d_ws;
    float* cscore = (float*)((char*)d_ws + cbytes);
    int*   cidx   = (int*)(cscore + (size_t)NPAIR * (size_t)cap);

    k_zero_counters<<<(NPAIR + 255) / 256, 256, 0, stream>>>(counters);
    k_softmax_cand<<<(B_ * N_) / 128, 256, 0, stream>>>(conf, counters, cscore, cidx, cap);
    k_select_nms<<<NPAIR, 256, 0, stream>>>(counters, cscore, cidx, loc, anchors, out, cap);
}